// NNDModule_73040213835933
// MI455X (gfx1250) — compile-verified
//
#include <hip/hip_runtime.h>
#include <float.h>

typedef __attribute__((ext_vector_type(2))) float v2f;
typedef __attribute__((ext_vector_type(8))) float v8f;

#define BATCH 16
#define NPTS  4096
#define MPTS  4096
#define MTILES (MPTS / 16)          // 256 column tiles of 16
#define THREADS 256                 // 8 wave32s
#define ROWS_PER_BLOCK 128          // 8 waves * 16 rows

// D = A x B + C with V_WMMA_F32_16X16X4_F32.
// A row i (point p1_i): [x1, y1, z1, sq1]
// B col j (point p2_j): [-2x2, -2y2, -2z2, 1]
// => D[i][j] = sq1_i - 2 * dot(p1_i, p2_j); full dist = D[i][j] + sq2_j.
__global__ __launch_bounds__(THREADS) void chamfer_wmma_kernel(
    const float* __restrict__ p1,   // [B, N, 3]
    const float* __restrict__ p2,   // [B, M, 3]
    float* __restrict__ out)        // [B*N] dist1 then [B*M] dist2
{
    // B fragments pre-split by half-wave:
    //   s_frag[q]        = (-2x2, -2y2)  -> K0,K1 for lanes 0-15
    //   s_frag[MPTS + q] = (-2z2,  1.0)  -> K2,K3 for lanes 16-31
    __shared__ float2 s_frag[2 * MPTS];   // 64 KB
    __shared__ float  s_sq2[MPTS];        // 16 KB
    __shared__ float  s_d2[MPTS];         // 16 KB: per-block dist2 partials

    const int tid  = threadIdx.x;
    const int lane = tid & 31;
    const int wave = tid >> 5;
    const int col  = lane & 15;
    const bool hi  = (lane >= 16);     // upper half-wave: rows +8, K=2..3
    const int b    = blockIdx.y;
    const int nblk = blockIdx.x;

    // Phase 0: build split B fragments for the whole batch of points2.
    const float* gp2 = p2 + (size_t)b * MPTS * 3;
    for (int i = tid; i < MPTS; i += THREADS) {
        const float x2 = gp2[i * 3 + 0];
        const float y2 = gp2[i * 3 + 1];
        const float z2 = gp2[i * 3 + 2];
        s_frag[i]        = make_float2(-2.0f * x2, -2.0f * y2);
        s_frag[MPTS + i] = make_float2(-2.0f * z2, 1.0f);
        s_sq2[i] = x2 * x2 + y2 * y2 + z2 * z2;
        s_d2[i]  = FLT_MAX;
    }
    __syncthreads();

    // Build A fragment: 16 points of set1 owned by this wave.
    const int n0 = nblk * ROWS_PER_BLOCK + wave * 16;
    const float* gp1 = p1 + ((size_t)b * NPTS + n0 + col) * 3;
    const float x1 = gp1[0], y1 = gp1[1], z1 = gp1[2];
    const float sq1 = x1 * x1 + y1 * y1 + z1 * z1;
    v2f afrag;
    afrag.x = hi ? z1  : x1;   // K=0 (lo) / K=2 (hi)
    afrag.y = hi ? sq1 : y1;   // K=1 (lo) / K=3 (hi)

    const int fbase = (hi ? MPTS : 0) + col;   // per-lane fragment base index

    v8f rowmin;
#pragma unroll
    for (int r = 0; r < 8; ++r) rowmin[r] = FLT_MAX;

    for (int mt = 0; mt < MTILES; ++mt) {
        const int q = mt * 16 + col;
        const float2 fr = s_frag[fbase + mt * 16];  // ds_load_b64, no select
        const float sq2 = s_sq2[q];                 // ds_load_b32
        v2f bfrag;
        bfrag.x = fr.x;
        bfrag.y = fr.y;

        v8f c = {};
        c = __builtin_amdgcn_wmma_f32_16x16x4_f32(
                /*neg_a=*/false, afrag, /*neg_b=*/false, bfrag,
                /*c_mod=*/(short)0, c, /*reuse_a=*/false, /*reuse_b=*/false);

        // c[r] = sq1 - 2*cross for row (r + 8*hi), column q; add sq2_q.
        float cmin = FLT_MAX;
#pragma unroll
        for (int r = 0; r < 8; ++r) {
            const float d = c[r] + sq2;
            rowmin[r] = fminf(rowmin[r], d);
            cmin = fminf(cmin, d);
        }
        // Both half-waves atomically fp-min their 8-row partial into the
        // column slot; no cross-lane merge or exec masking needed.
        __hip_atomic_fetch_min(&s_d2[q], cmin, __ATOMIC_RELAXED,
                               __HIP_MEMORY_SCOPE_WORKGROUP);
    }

    // Butterfly row-min across the 16 columns (stays within each half-wave).
#pragma unroll
    for (int m = 8; m >= 1; m >>= 1) {
#pragma unroll
        for (int r = 0; r < 8; ++r)
            rowmin[r] = fminf(rowmin[r], __shfl_xor(rowmin[r], m, 32));
    }
    if (col == 0) {
        const int nbase = n0 + (hi ? 8 : 0);
        float* o1 = out + (size_t)b * NPTS + nbase;
#pragma unroll
        for (int r = 0; r < 8; ++r) o1[r] = rowmin[r];
    }

    __syncthreads();
    // Merge this block's dist2 partials into global with device-scope fp-min.
    float* o2 = out + (size_t)BATCH * NPTS + (size_t)b * MPTS;
    for (int i = tid; i < MPTS; i += THREADS) {
        __hip_atomic_fetch_min(&o2[i], s_d2[i], __ATOMIC_RELAXED,
                               __HIP_MEMORY_SCOPE_AGENT);
    }
}

__global__ __launch_bounds__(THREADS) void init_dist2_kernel(float* out) {
    const int i = blockIdx.x * THREADS + threadIdx.x;
    out[(size_t)BATCH * NPTS + i] = FLT_MAX;
}

extern "C" void kernel_launch(void* const* d_in, const int* in_sizes, int n_in,
                              void* d_out, int out_size, void* d_ws, size_t ws_size,
                              hipStream_t stream) {
    (void)in_sizes; (void)n_in; (void)out_size; (void)d_ws; (void)ws_size;
    const float* p1 = (const float*)d_in[0];
    const float* p2 = (const float*)d_in[1];
    float* out = (float*)d_out;

    // Seed the dist2 half of the output with +MAX each launch (atomic-min target).
    init_dist2_kernel<<<(BATCH * MPTS) / THREADS, THREADS, 0, stream>>>(out);

    dim3 grid(NPTS / ROWS_PER_BLOCK, BATCH);   // (32, 16)
    chamfer_wmma_kernel<<<grid, THREADS, 0, stream>>>(p1, p2, out);
}